// LaplacianKnn_66606352827344
// MI455X (gfx1250) — compile-verified
//
#include <hip/hip_runtime.h>

// ---------------------------------------------------------------------------
// kNN-graph Matern quadratic form  x^T L^3 x  on MI455X (gfx1250, wave32).
//
// Strategy (memory-roofline driven):
//   K1: D[i]       = sum_k exp(-dist[i,k]/eps)                 (stream 128MB NT)
//   K2: vals[i,k]  = -(4/eps) * u_k / sum(u),  u_k = e_k/D[j]  (D_i cancels)
//   K3: y <- diag*y + sum_k vals[i,k]*y[idx[i,k]]   (x -> y0 -> y1)
//   K4: fused 3rd SpMV + x_i*y3_i, reduced per-wave with fp32 WMMA
//       (B = ones => D[m][n] = rowsum_m(A), exact in f32), block partials
//       to workspace, deterministic single-block final reduce (also WMMA).
//
// Streams use non-temporal loads so the 4MB gather targets (y, D) stay
// resident in the 192MB L2; random gathers then hit L2, not HBM.
// ---------------------------------------------------------------------------

typedef __attribute__((ext_vector_type(2))) float v2f;
typedef __attribute__((ext_vector_type(4))) float v4f;
typedef __attribute__((ext_vector_type(8))) float v8f;
typedef __attribute__((ext_vector_type(4))) int   v4i;

#define KNN   32
#define BLOCK 256
#define NWAVE (BLOCK / 32)

// Exact fp32 reduction of the 32 lane values `p` of this wave using
// V_WMMA_F32_16X16X4_F32 with B == ones:
//   D[m][n] = sum_k A[m][k]  (independent of B's VGPR layout).
// Stage 1: A[m] = [p_m, 0, p_{m+16}, 0]  -> t_m = p_m + p_{m+16}, replicated.
// Stage 2: 4 accumulating WMMAs over pairs (d_j, d_{j+1}) sum all 16 t_m.
// Requires EXEC == all ones (callers ensure no early-out before this point).
__device__ __forceinline__ float wave_reduce_wmma_f32(float p) {
  v2f ones; ones[0] = 1.0f; ones[1] = 1.0f;
  v2f a;    a[0] = p;       a[1] = 0.0f;
  v8f z = {};
  v8f t = __builtin_amdgcn_wmma_f32_16x16x4_f32(false, a, false, ones,
                                                (short)0, z, false, false);
  v8f c = {};
  v2f a2;
  a2[0] = t[0]; a2[1] = t[1];
  c = __builtin_amdgcn_wmma_f32_16x16x4_f32(false, a2, false, ones,
                                            (short)0, c, false, false);
  a2[0] = t[2]; a2[1] = t[3];
  c = __builtin_amdgcn_wmma_f32_16x16x4_f32(false, a2, false, ones,
                                            (short)0, c, false, false);
  a2[0] = t[4]; a2[1] = t[5];
  c = __builtin_amdgcn_wmma_f32_16x16x4_f32(false, a2, false, ones,
                                            (short)0, c, false, false);
  a2[0] = t[6]; a2[1] = t[7];
  c = __builtin_amdgcn_wmma_f32_16x16x4_f32(false, a2, false, ones,
                                            (short)0, c, false, false);
  // C/D layout: VGPR0, lanes 0-15 hold row M=0; every element == total.
  return c[0];
}

// K1: D[i] = sum_k exp(-dist[i,k] / eps)
__global__ void k_rowsum_D(const float* __restrict__ dist,
                           const float* __restrict__ eps_p,
                           float* __restrict__ D, int N) {
  int i = blockIdx.x * BLOCK + threadIdx.x;
  if (i >= N) return;
  float ie = 1.0f / eps_p[0];
  const v4f* row = (const v4f*)(dist + (size_t)i * KNN);
  float s = 0.0f;
#pragma unroll
  for (int q = 0; q < KNN / 4; ++q) {
    v4f d = __builtin_nontemporal_load(row + q);
    s += __expf(-d[0] * ie) + __expf(-d[1] * ie) +
         __expf(-d[2] * ie) + __expf(-d[3] * ie);
  }
  D[i] = s;
}

// K2: vals[i,k] = -(4/eps) * u_k / S,  u_k = exp(-d_k/eps)/D[idx_k], S = sum u.
// (The reference also divides by D_i, but D_i cancels in the row normalize.)
__global__ void k_build_vals(const float* __restrict__ dist,
                             const int*   __restrict__ idx,
                             const float* __restrict__ D,
                             const float* __restrict__ eps_p,
                             float* __restrict__ vals, int N) {
  int i = blockIdx.x * BLOCK + threadIdx.x;
  if (i >= N) return;
  float ie = 1.0f / eps_p[0];
  const v4f* drow = (const v4f*)(dist + (size_t)i * KNN);
  const v4i* irow = (const v4i*)(idx  + (size_t)i * KNN);
  float u[KNN];
  float S = 0.0f;
#pragma unroll
  for (int q = 0; q < KNN / 4; ++q) {
    v4f d  = __builtin_nontemporal_load(drow + q);
    v4i id = __builtin_nontemporal_load(irow + q);
#pragma unroll
    for (int j = 0; j < 4; ++j) {
      float e = __expf(-d[j] * ie) / D[id[j]];
      u[4 * q + j] = e;
      S += e;
    }
  }
  float scale = -4.0f * ie / S;
  v4f* vrow = (v4f*)(vals + (size_t)i * KNN);
#pragma unroll
  for (int q = 0; q < KNN / 4; ++q) {
    v4f o;
#pragma unroll
    for (int j = 0; j < 4; ++j) o[j] = u[4 * q + j] * scale;
    __builtin_nontemporal_store(o, vrow + q);
  }
}

// K3/K4: one Laplacian application. PRECOMP selects precomputed vals vs.
// on-the-fly recompute (workspace-size fallback). FUSE_DOT replaces the
// y-store with p = x_i * y_i and a WMMA wave reduction -> block partial.
template <bool PRECOMP, bool FUSE_DOT>
__global__ void k_spmv(const float* __restrict__ yin,
                       const int*   __restrict__ idx,
                       const float* __restrict__ vals,
                       const float* __restrict__ dist,
                       const float* __restrict__ D,
                       const float* __restrict__ eps_p,
                       const float* __restrict__ kk_p,
                       const int*   __restrict__ nu_p,
                       const float* __restrict__ x,
                       float* __restrict__ yout,
                       float* __restrict__ partials, int N) {
  __shared__ float wsum[NWAVE];
  int i = blockIdx.x * BLOCK + threadIdx.x;
  float ie = 1.0f / eps_p[0];
  float kk = kk_p[0];
  float diag = 4.0f * ie + 2.0f * (float)nu_p[0] / (kk * kk) + 10.0f;

  float p = 0.0f;
  if (i < N) {  // no early return: WMMA below needs EXEC all-ones
    const v4i* irow = (const v4i*)(idx + (size_t)i * KNN);
    float acc = diag * yin[i];
    if (PRECOMP) {
      const v4f* vrow = (const v4f*)(vals + (size_t)i * KNN);
#pragma unroll
      for (int q = 0; q < KNN / 4; ++q) {
        v4i id = __builtin_nontemporal_load(irow + q);
        v4f v  = __builtin_nontemporal_load(vrow + q);
        acc += v[0] * yin[id[0]] + v[1] * yin[id[1]] +
               v[2] * yin[id[2]] + v[3] * yin[id[3]];
      }
    } else {
      const v4f* drow = (const v4f*)(dist + (size_t)i * KNN);
      float u[KNN]; int id[KNN];
      float S = 0.0f;
#pragma unroll
      for (int q = 0; q < KNN / 4; ++q) {
        v4f d  = __builtin_nontemporal_load(drow + q);
        v4i iv = __builtin_nontemporal_load(irow + q);
#pragma unroll
        for (int j = 0; j < 4; ++j) {
          int t = iv[j];
          float e = __expf(-d[j] * ie) / D[t];
          u[4 * q + j] = e;
          id[4 * q + j] = t;
          S += e;
        }
      }
      float scale = -4.0f * ie / S;
#pragma unroll
      for (int j = 0; j < KNN; ++j) acc += (u[j] * scale) * yin[id[j]];
    }
    if (FUSE_DOT) p = x[i] * acc;
    else          yout[i] = acc;
  }

  if (FUSE_DOT) {
    float tot = wave_reduce_wmma_f32(p);
    int lane = threadIdx.x & 31;
    int wv   = threadIdx.x >> 5;
    if (lane == 0) wsum[wv] = tot;
    __syncthreads();
    if (threadIdx.x == 0) {
      float s = 0.0f;
#pragma unroll
      for (int j = 0; j < NWAVE; ++j) s += wsum[j];
      partials[blockIdx.x] = s;
    }
  }
}

// Deterministic final reduction of per-block partials (single block).
__global__ void k_final_reduce(const float* __restrict__ partials, int nb,
                               float* __restrict__ out) {
  __shared__ float wsum[NWAVE];
  float p = 0.0f;
  for (int j = threadIdx.x; j < nb; j += BLOCK) p += partials[j];
  float tot = wave_reduce_wmma_f32(p);  // all 256 threads converged here
  int lane = threadIdx.x & 31;
  int wv   = threadIdx.x >> 5;
  if (lane == 0) wsum[wv] = tot;
  __syncthreads();
  if (threadIdx.x == 0) {
    float s = 0.0f;
#pragma unroll
    for (int j = 0; j < NWAVE; ++j) s += wsum[j];
    out[0] = s;
  }
}

extern "C" void kernel_launch(void* const* d_in, const int* in_sizes, int n_in,
                              void* d_out, int out_size, void* d_ws,
                              size_t ws_size, hipStream_t stream) {
  const float* x    = (const float*)d_in[0];
  const int*   idx  = (const int*)  d_in[1];
  const float* dist = (const float*)d_in[2];
  const float* eps  = (const float*)d_in[3];
  const float* kk   = (const float*)d_in[4];
  const int*   nu   = (const int*)  d_in[5];   // NU == 3 (static in reference)
  int N = in_sizes[0];
  float* out = (float*)d_out;

  int nb = (N + BLOCK - 1) / BLOCK;

  // Workspace layout (floats): D[N] | y0[N] | y1[N] | partials[8192] | vals[N*K]
  float* ws       = (float*)d_ws;
  float* D        = ws;
  float* y0       = ws + (size_t)N;
  float* y1       = ws + (size_t)2 * N;
  float* partials = ws + (size_t)3 * N;
  float* vals     = ws + (size_t)3 * N + 8192;
  size_t need_full =
      ((size_t)3 * N + 8192 + (size_t)N * KNN) * sizeof(float);
  bool precomp = (ws_size >= need_full);

  k_rowsum_D<<<nb, BLOCK, 0, stream>>>(dist, eps, D, N);

  if (precomp) {
    k_build_vals<<<nb, BLOCK, 0, stream>>>(dist, idx, D, eps, vals, N);
    k_spmv<true, false><<<nb, BLOCK, 0, stream>>>(
        x,  idx, vals, dist, D, eps, kk, nu, x, y0, nullptr, N);
    k_spmv<true, false><<<nb, BLOCK, 0, stream>>>(
        y0, idx, vals, dist, D, eps, kk, nu, x, y1, nullptr, N);
    k_spmv<true, true><<<nb, BLOCK, 0, stream>>>(
        y1, idx, vals, dist, D, eps, kk, nu, x, nullptr, partials, N);
  } else {
    k_spmv<false, false><<<nb, BLOCK, 0, stream>>>(
        x,  idx, nullptr, dist, D, eps, kk, nu, x, y0, nullptr, N);
    k_spmv<false, false><<<nb, BLOCK, 0, stream>>>(
        y0, idx, nullptr, dist, D, eps, kk, nu, x, y1, nullptr, N);
    k_spmv<false, true><<<nb, BLOCK, 0, stream>>>(
        y1, idx, nullptr, dist, D, eps, kk, nu, x, nullptr, partials, N);
  }

  k_final_reduce<<<1, BLOCK, 0, stream>>>(partials, nb, out);
}